// AdaAT_49117245997616
// MI455X (gfx1250) — compile-verified
//
#include <hip/hip_runtime.h>
#include <math.h>

typedef float v2f __attribute__((ext_vector_type(2)));
typedef float v8f __attribute__((ext_vector_type(8)));
typedef unsigned int v4u __attribute__((ext_vector_type(4)));
typedef unsigned int v8u __attribute__((ext_vector_type(8)));

#define PI_F 3.14159f

// ---------------------------------------------------------------------------
// Kernel 1: parameter MLP via f32 WMMA (single workgroup, 8 waves)
//   p     = relu(para @ W1 + b1)                       (8 x 256)
//   scale = 2*sigmoid(p @ Ws + bs)                     (8 x 256)
//   angle = tanh(p @ Wr + br) * PI  -> cos, sin        (8 x 256)
//   trans = tanh(p @ Wt + bt)                          (8 x 256 x 2)
// Output layout in d_ws (floats, idx = b*256 + d):
//   [0,2048) scale | [2048,4096) cos | [4096,6144) sin | [6144,8192) tx | [8192,10240) ty
// ---------------------------------------------------------------------------
__launch_bounds__(256)
__global__ void adaat_params_wmma(const float* __restrict__ para,
                                  const float* __restrict__ W1, const float* __restrict__ b1,
                                  const float* __restrict__ Ws, const float* __restrict__ bs,
                                  const float* __restrict__ Wr, const float* __restrict__ br,
                                  const float* __restrict__ Wt, const float* __restrict__ bt,
                                  float* __restrict__ out) {
  __shared__ float a_lds[16 * 256];   // para_code, rows 8..15 zero
  __shared__ float p_lds[16 * 256];   // hidden p,  rows 8..15 zero
  const int tid = threadIdx.x;

  for (int i = tid; i < 16 * 256; i += 256) { a_lds[i] = 0.0f; p_lds[i] = 0.0f; }
  __syncthreads();
  for (int i = tid; i < 8 * 256; i += 256) a_lds[i] = para[i];
  __syncthreads();

  const int lane  = tid & 31;
  const int wave  = tid >> 5;
  const int row16 = lane & 15;         // A row / B,D column within tile
  const int khalf = (lane >> 4) << 1;  // 0 or 2 : K sub-slot for A/B fragments
  const int rbase = (lane >> 4) << 3;  // 0 or 8 : D row base (lanes 16..31 -> rows 8..15)

  // ---- GEMM 1: p = relu(para @ W1 + b1), 16 output tiles of 16 cols ----
  for (int t = wave; t < 16; t += 8) {
    const int n0 = t * 16;
    v8f acc = {};
    for (int k = 0; k < 256; k += 4) {
      const int kk = k + khalf;
      v2f a, b;
      a.x = a_lds[row16 * 256 + kk];
      a.y = a_lds[row16 * 256 + kk + 1];
      b.x = W1[kk * 256 + n0 + row16];
      b.y = W1[(kk + 1) * 256 + n0 + row16];
      acc = __builtin_amdgcn_wmma_f32_16x16x4_f32(false, a, false, b,
                                                  (short)0, acc, false, false);
    }
    if (rbase == 0) {               // lanes 0..15 hold batch rows 0..7
      const int col = n0 + row16;
      const float bb = b1[col];
      for (int v = 0; v < 8; ++v) {
        p_lds[v * 256 + col] = fmaxf(acc[v] + bb, 0.0f);
      }
    }
  }
  __syncthreads();

  // ---- GEMMs 2..4 fused: 64 tiles across [Ws | Wr | Wt] ----
  for (int t = wave; t < 64; t += 8) {
    const float* Wm; const float* bias; int stride; int n0;
    if (t < 16)      { Wm = Ws; bias = bs; stride = 256; n0 = t * 16; }
    else if (t < 32) { Wm = Wr; bias = br; stride = 256; n0 = (t - 16) * 16; }
    else             { Wm = Wt; bias = bt; stride = 512; n0 = (t - 32) * 16; }

    v8f acc = {};
    for (int k = 0; k < 256; k += 4) {
      const int kk = k + khalf;
      v2f a, b;
      a.x = p_lds[row16 * 256 + kk];
      a.y = p_lds[row16 * 256 + kk + 1];
      b.x = Wm[kk * stride + n0 + row16];
      b.y = Wm[(kk + 1) * stride + n0 + row16];
      acc = __builtin_amdgcn_wmma_f32_16x16x4_f32(false, a, false, b,
                                                  (short)0, acc, false, false);
    }
    if (rbase == 0) {
      const int col = n0 + row16;
      const float bb = bias[col];
      for (int v = 0; v < 8; ++v) {
        const float val = acc[v] + bb;
        if (t < 16) {
          out[v * 256 + col] = 2.0f / (1.0f + expf(-val));          // scale
        } else if (t < 32) {
          const float ang = tanhf(val) * PI_F;
          out[2048 + v * 256 + col] = cosf(ang);
          out[4096 + v * 256 + col] = sinf(ang);
        } else {
          const float tv = tanhf(val);
          const int d = col >> 1;                                    // (d, comp)
          out[6144 + ((col & 1) << 11) + v * 256 + d] = tv;          // tx / ty
        }
      }
    }
  }
}

// ---------------------------------------------------------------------------
// Kernel 2: per-(b,d) affine grid_sample.
// The 64 KB source image is DMA'd into LDS by the Tensor Data Mover:
// wave 0 issues one tensor_load_to_lds with a 2D D# (128x128, 4B elements),
// waits on TENSORcnt, then all 8 waves sample from LDS.
// ---------------------------------------------------------------------------
__launch_bounds__(256)
__global__ void adaat_sample(const float* __restrict__ fmap,
                             const float* __restrict__ prm,
                             float* __restrict__ out) {
  __shared__ float img[128 * 128];
  const int bd = blockIdx.x;                     // 0..2047

  const float sc = prm[bd];
  const float c  = prm[2048 + bd];
  const float s  = prm[4096 + bd];
  const float tx = prm[6144 + bd];
  const float ty = prm[8192 + bd];

  if (threadIdx.x < 32) {
    // ---- Tensor DMA descriptor (ISA 8.3/8.4), all wave-uniform ----
    const unsigned long long gaddr =
        (unsigned long long)(const void*)(fmap + (size_t)bd * 16384u);
    const unsigned int lds_base =
        (unsigned int)(unsigned long long)(const void*)img;  // low 32b of LDS aperture = LDS offset

    v4u g0;
    g0.x = 1u;                                   // count=1 (valid user descriptor)
    g0.y = lds_base;                             // lds_addr (bytes)
    g0.z = (unsigned int)gaddr;                  // global_addr[31:0]
    g0.w = ((unsigned int)(gaddr >> 32) & 0x01FFFFFFu) | (2u << 30); // addr[56:32] | type=2

    v8u g1;
    g1.s0 = (2u << 16);      // workgroup_mask=0, data_size=2 (4 bytes), no flags
    g1.s1 = (128u << 16);    // atomic_barrier_addr=0 | tensor_dim0[15:0]=128
    g1.s2 = (128u << 16);    // tensor_dim0[31:16]=0  | tensor_dim1[15:0]=128
    g1.s3 = (128u << 16);    // tensor_dim1[31:16]=0  | tile_dim0=128
    g1.s4 = 128u;            // tile_dim1=128 | tile_dim2=0
    g1.s5 = 128u;            // tensor_dim0_stride[31:0]=128 (elements)
    g1.s6 = 0u;              // tensor_dim0_stride[47:32]=0 | tensor_dim1_stride[15:0]=0
    g1.s7 = 0u;              // tensor_dim1_stride[47:16]=0

    asm volatile("tensor_load_to_lds %0, %1"
                 :
                 : "s"(g0), "s"(g1)
                 : "memory");
    __builtin_amdgcn_s_wait_tensorcnt(0);
  }
  __syncthreads();

  float* dst = out + (size_t)bd * 16384u;
  for (int i = threadIdx.x; i < 16384; i += 256) {
    const int px = i & 127;
    const int py = i >> 7;
    const float gx = (float)px * (2.0f / 127.0f) - 1.0f;
    const float gy = (float)py * (2.0f / 127.0f) - 1.0f;
    const float nx = (c * gx - s * gy) * sc + tx;
    const float ny = (s * gx + c * gy) * sc + ty;

    // unnormalize (align_corners=False) + border clamp
    float x = fminf(fmaxf((nx + 1.0f) * 64.0f - 0.5f, 0.0f), 127.0f);
    float y = fminf(fmaxf((ny + 1.0f) * 64.0f - 0.5f, 0.0f), 127.0f);
    const float x0f = floorf(x), y0f = floorf(y);
    const float wx = x - x0f,   wy = y - y0f;
    const int x0 = (int)x0f;
    const int y0 = (int)y0f;
    const int x1 = (x0 < 127) ? x0 + 1 : 127;
    const int y1 = (y0 < 127) ? y0 + 1 : 127;

    const float* r0 = img + (y0 << 7);
    const float* r1 = img + (y1 << 7);
    const float v00 = r0[x0], v01 = r0[x1], v10 = r1[x0], v11 = r1[x1];
    const float top = v00 + wx * (v01 - v00);
    const float bot = v10 + wx * (v11 - v10);
    dst[i] = top + wy * (bot - top);
  }
}

// ---------------------------------------------------------------------------
extern "C" void kernel_launch(void* const* d_in, const int* in_sizes, int n_in,
                              void* d_out, int out_size, void* d_ws, size_t ws_size,
                              hipStream_t stream) {
  const float* fmap = (const float*)d_in[0];
  const float* para = (const float*)d_in[1];
  const float* W1   = (const float*)d_in[2];
  const float* b1   = (const float*)d_in[3];
  const float* Ws   = (const float*)d_in[4];
  const float* bs   = (const float*)d_in[5];
  const float* Wr   = (const float*)d_in[6];
  const float* br   = (const float*)d_in[7];
  const float* Wt   = (const float*)d_in[8];
  const float* bt   = (const float*)d_in[9];
  float* prm = (float*)d_ws;   // 10240 floats of scratch

  adaat_params_wmma<<<1, 256, 0, stream>>>(para, W1, b1, Ws, bs, Wr, br, Wt, bt, prm);
  adaat_sample<<<2048, 256, 0, stream>>>(fmap, prm, (float*)d_out);
}